// DurationCalculator_87677462380947
// MI455X (gfx1250) — compile-verified
//
#include <hip/hip_runtime.h>

// Problem constants (match the reference module).
#define B_DIM        256
#define Y_DIM        4096
#define X_DIM        1024
#define MASK_PENALTY (-10000)
#define NTHREADS     256
#define TILE_INTS    (NTHREADS * 4)      // 1024 int32 per stage (4 KB)
#define NSTAGES      (Y_DIM / TILE_INTS) // 4

typedef int v4i __attribute__((ext_vector_type(4)));

// ---- CDNA5 async global->LDS path (gfx1250), guarded so we always compile ----
#if defined(__AMDGCN__) && __has_builtin(__builtin_amdgcn_global_load_async_to_lds_b128)
#define HAVE_ASYNC_LDS 1
#else
#define HAVE_ASYNC_LDS 0
#endif

#if HAVE_ASYNC_LDS
// Builtin takes pointers to 16-byte int vectors in global (AS1) / LDS (AS3).
typedef int gvec4i __attribute__((vector_size(16)));
typedef __attribute__((address_space(1))) gvec4i GA_v4i;   // global memory
typedef __attribute__((address_space(3))) gvec4i LDS_v4i;  // LDS
#endif

#if defined(__AMDGCN__) && __has_builtin(__builtin_amdgcn_s_wait_asynccnt)
#define WAIT_ASYNCCNT(n) __builtin_amdgcn_s_wait_asynccnt(n)
#elif defined(__AMDGCN__)
#define WAIT_ASYNCCNT(n) asm volatile("s_wait_asynccnt %0" ::"i"(n) : "memory")
#else
#define WAIT_ASYNCCNT(n)
#endif

__global__ __launch_bounds__(NTHREADS) void duration_calc_kernel(
    const int* __restrict__ duration,       // [B, Y] int32
    const int* __restrict__ output_length,  // [B]    int32
    int* __restrict__ wa_out,               // [B, Y] int32 (weights_argmax)
    int* __restrict__ dur_out)              // [B, X] int32 (durations)
{
    __shared__ alignas(16) int hist[X_DIM]; // 4 KB per-batch histogram
#if HAVE_ASYNC_LDS
    __shared__ alignas(16) int tile[2][TILE_INTS]; // 8 KB double buffer
#endif

    const int tid = threadIdx.x;
    const int b   = blockIdx.x;
    const int len = output_length[b];
    const int off = tid * 4;                // this lane's 16B slot (1024/256 = 4 ints)

    const int* gsrc = duration + (size_t)b * Y_DIM;
    int* wa_dst = wa_out + (size_t)b * Y_DIM;

    // Zero the histogram: exactly one ds_store_b128 per lane.
    *(v4i*)&hist[off] = (v4i)0;

#if HAVE_ASYNC_LDS
    // Prime the pipeline: async-load stage 0 into LDS buffer 0 (ASYNCcnt++).
    __builtin_amdgcn_global_load_async_to_lds_b128(
        (GA_v4i*)(gsrc + off), (LDS_v4i*)&tile[0][off], 0, 0);
#endif

    __syncthreads();  // hist zeros visible before any ds_add

#if HAVE_ASYNC_LDS
    #pragma unroll
    for (int s = 0; s < NSTAGES; ++s) {
        if (s + 1 < NSTAGES) {
            // Issue next stage, then wait until only it is outstanding
            // (async loads complete in order -> stage s is done in LDS).
            __builtin_amdgcn_global_load_async_to_lds_b128(
                (GA_v4i*)(gsrc + (s + 1) * TILE_INTS + off),
                (LDS_v4i*)&tile[(s + 1) & 1][off], 0, 0);
            WAIT_ASYNCCNT(1);
        } else {
            WAIT_ASYNCCNT(0);
        }

        const v4i d      = *(const v4i*)&tile[s & 1][off];  // ds_load_b128, own slot
        const int base_y = s * TILE_INTS + off;

        v4i wa;
        #pragma unroll
        for (int j = 0; j < 4; ++j) {
            const int y = base_y + j;
            const int w = d[j] + ((y < len) ? 0 : MASK_PENALTY);
            wa[j] = w;
            // Reference computes bin X for out-of-range then drops it -> skip.
            if ((unsigned)w < (unsigned)X_DIM) atomicAdd(&hist[w], 1);  // ds_add_u32
        }
        // Streaming output: never re-read, keep it out of near caches.
        __builtin_nontemporal_store(wa, (v4i*)(wa_dst + base_y));
    }
#else
    // Fallback: direct coalesced b128 global loads (also used for the host pass).
    for (int s = 0; s < NSTAGES; ++s) {
        const int base_y = s * TILE_INTS + off;
        const v4i d = *(const v4i*)(gsrc + base_y);
        v4i wa;
        #pragma unroll
        for (int j = 0; j < 4; ++j) {
            const int y = base_y + j;
            const int w = d[j] + ((y < len) ? 0 : MASK_PENALTY);
            wa[j] = w;
            if ((unsigned)w < (unsigned)X_DIM) atomicAdd(&hist[w], 1);
        }
        __builtin_nontemporal_store(wa, (v4i*)(wa_dst + base_y));
    }
#endif

    __syncthreads();  // all ds_add done before reading bins

    // 1024 bins / 256 threads = one b128 per lane, coalesced, non-temporal.
    const v4i h = *(const v4i*)&hist[off];
    __builtin_nontemporal_store(h, (v4i*)(dur_out + (size_t)b * X_DIM + off));
}

extern "C" void kernel_launch(void* const* d_in, const int* in_sizes, int n_in,
                              void* d_out, int out_size, void* d_ws, size_t ws_size,
                              hipStream_t stream) {
    (void)in_sizes; (void)n_in; (void)d_ws; (void)ws_size; (void)out_size;

    const int* duration      = (const int*)d_in[0];  // [B, Y] int32
    const int* output_length = (const int*)d_in[1];  // [B]    int32
    // d_in[2] is x_steps (scalar int, == X_DIM); fixed at compile time here.

    int* out     = (int*)d_out;
    int* wa_out  = out;                              // B*Y elements
    int* dur_out = out + (size_t)B_DIM * Y_DIM;      // B*X elements

    duration_calc_kernel<<<B_DIM, NTHREADS, 0, stream>>>(
        duration, output_length, wa_out, dur_out);
}